// FS_Sampler_54898271977793
// MI455X (gfx1250) — compile-verified
//
#include <hip/hip_runtime.h>

#define NPTS    8192
#define NBATCH  2
#define NPOINT  2048
#define NCHAN   64
#define R2      0.04f   // R=0.2

typedef __attribute__((ext_vector_type(2))) float v2f;
typedef __attribute__((ext_vector_type(8))) float v8f;

// ---------------------------------------------------------------------------
// Density kernel: per point i, count j with ||pi-pj||^2 < R^2.
// Gram matrix via V_WMMA_F32_16X16X4_F32 (K=4: x,y,z,0).
// One wave handles one 16-row tile of one batch and sweeps all 512 col tiles.
// Total waves = NBATCH * (NPTS/16) = 1024 -> 128 blocks x 256 threads.
// ---------------------------------------------------------------------------
__global__ __launch_bounds__(256) void density_kernel(
    const float* __restrict__ pts,   // [B][N][3]
    float* __restrict__ dens)        // [B][N]
{
  const int gwave = (blockIdx.x * blockDim.x + threadIdx.x) >> 5;
  const int lane  = threadIdx.x & 31;
  const int b     = gwave >> 9;            // 512 row-tiles per batch
  const int itile = gwave & 511;
  const int i0    = itile * 16;
  const int hh    = lane >> 4;             // half of the wave (K split)
  const int n     = lane & 15;
  const float* P  = pts + (size_t)b * NPTS * 3;

  // A-matrix 16x4 f32 layout: lanes 0-15: VGPR0=K0, VGPR1=K1 ; lanes 16-31: K2, K3
  const float* prow = P + (size_t)(i0 + n) * 3;
  v2f a;
  a.x = hh ? prow[2] : prow[0];
  a.y = hh ? 0.0f    : prow[1];

  // squared norms of the 8 rows this lane's D-VGPRs correspond to (M = 8*hh + v)
  float aar[8];
#pragma unroll
  for (int v = 0; v < 8; ++v) {
    const float* pr = P + (size_t)(i0 + hh * 8 + v) * 3;
    aar[v] = pr[0]*pr[0] + pr[1]*pr[1] + pr[2]*pr[2];
  }

  float cnt[8];
#pragma unroll
  for (int v = 0; v < 8; ++v) cnt[v] = 0.0f;

  for (int jt = 0; jt < NPTS / 16; ++jt) {
    const int col   = jt * 16 + n;
    const float* pc = P + (size_t)col * 3;
    const float cx = pc[0], cy = pc[1], cz = pc[2];
    // B-matrix 4x16 f32: mirror of A layout (rows K striped across lanes)
    v2f bmat;
    bmat.x = hh ? cz   : cx;
    bmat.y = hh ? 0.0f : cy;
    const float aac = cx*cx + cy*cy + cz*cz;

    v8f c = {};
    v8f d = __builtin_amdgcn_wmma_f32_16x16x4_f32(
        /*neg_a=*/false, a, /*neg_b=*/false, bmat,
        /*c_mod=*/(short)0, c, /*reuse_a=*/false, /*reuse_b=*/false);

    // D element (M = 8*hh + v, Ncol = n) in VGPR v of this lane
#pragma unroll
    for (int v = 0; v < 8; ++v) {
      const float dist = aar[v] + aac - 2.0f * d[v];
      cnt[v] += (dist < R2) ? 1.0f : 0.0f;
    }
  }

  // sum counts across the 16 lanes of each half (xor masks < 16 stay in-half)
#pragma unroll
  for (int off = 1; off < 16; off <<= 1) {
#pragma unroll
    for (int v = 0; v < 8; ++v)
      cnt[v] += __shfl_xor(cnt[v], off, 32);
  }
  if (n == 0) {
#pragma unroll
    for (int v = 0; v < 8; ++v)
      dens[(size_t)b * NPTS + i0 + hh * 8 + v] = cnt[v];
  }
}

// ---------------------------------------------------------------------------
// FPS kernel: one workgroup per (batch, variant). variant 0 = F-FPS (67-dim
// fused space), variant 1 = D-FPS (eucl^2 + weighted manhattan, density
// weighted). 1024 threads, 8 points/thread, min_d kept in registers.
// ---------------------------------------------------------------------------
__global__ __launch_bounds__(1024) void fps_kernel(
    const float* __restrict__ pts,    // [B][N][3]
    const float* __restrict__ feats,  // [B][C][N]
    const float* __restrict__ dens,   // [B][N]
    int* __restrict__ out)            // [B][2*NPOINT]
{
  const int b       = blockIdx.x >> 1;
  const int variant = blockIdx.x & 1;      // 0 = F-FPS, 1 = D-FPS
  const int tid     = threadIdx.x;
  const int jbase   = tid * 8;
  const float* P = pts   + (size_t)b * NPTS * 3;
  const float* F = feats + (size_t)b * NCHAN * NPTS;
  int* ob = out + (size_t)b * (2 * NPOINT) + (size_t)variant * NPOINT;

  __shared__ float sVec[3 + NCHAN];
  __shared__ float wval[32];
  __shared__ int   widx[32];
  __shared__ int   sLast;

  float min_d[8], w[8];
#pragma unroll
  for (int p = 0; p < 8; ++p) {
    min_d[p] = 1e10f;
    w[p] = variant ? (1.0f / dens[(size_t)b * NPTS + jbase + p]) : 1.0f;
  }

  if (tid == 0) { ob[0] = 0; sLast = 0; }
  __syncthreads();

  int last = 0;

  for (int s = 1; s < NPOINT; ++s) {
    // stage the most recently selected point's vector in LDS
    if (tid < 3)
      sVec[tid] = P[(size_t)last * 3 + tid];
    else if (variant == 0 && tid < 3 + NCHAN)
      sVec[tid] = F[(size_t)(tid - 3) * NPTS + last];
    __syncthreads();

    const float sx = sVec[0], sy = sVec[1], sz = sVec[2];

    float acc[8];
#pragma unroll
    for (int p = 0; p < 8; ++p) {
      const float* pp = P + (size_t)(jbase + p) * 3;
      const float dx = pp[0] - sx, dy = pp[1] - sy, dz = pp[2] - sz;
      const float e2 = dx*dx + dy*dy + dz*dz;
      acc[p] = variant ? (e2 + fabsf(dx) + fabsf(dy) + 2.0f * fabsf(dz)) : e2;
    }

    if (variant == 0) {
      // 64 feature channels, two float4 loads per channel (8 consecutive pts)
      for (int c = 0; c < NCHAN; ++c) {
        const float sc = sVec[3 + c];
        const float4* f4 =
            reinterpret_cast<const float4*>(F + (size_t)c * NPTS + jbase);
        const float4 fa = f4[0], fb = f4[1];
        float d0 = fa.x - sc, d1 = fa.y - sc, d2 = fa.z - sc, d3 = fa.w - sc;
        float d4 = fb.x - sc, d5 = fb.y - sc, d6 = fb.z - sc, d7 = fb.w - sc;
        acc[0] += d0*d0; acc[1] += d1*d1; acc[2] += d2*d2; acc[3] += d3*d3;
        acc[4] += d4*d4; acc[5] += d5*d5; acc[6] += d6*d6; acc[7] += d7*d7;
      }
    }

    // local best (argmax of min_d * w, ties -> lowest index)
    float bestv = -1.0f; int besti = 0;
#pragma unroll
    for (int p = 0; p < 8; ++p) {
      min_d[p] = fminf(min_d[p], acc[p]);
      const float sc = min_d[p] * w[p];
      if (sc > bestv) { bestv = sc; besti = jbase + p; }
    }
    // wave32 butterfly argmax
#pragma unroll
    for (int off = 16; off > 0; off >>= 1) {
      const float ov = __shfl_xor(bestv, off, 32);
      const int   oi = __shfl_xor(besti, off, 32);
      if (ov > bestv || (ov == bestv && oi < besti)) { bestv = ov; besti = oi; }
    }
    if ((tid & 31) == 0) { wval[tid >> 5] = bestv; widx[tid >> 5] = besti; }
    __syncthreads();
    if (tid < 32) {
      bestv = wval[tid]; besti = widx[tid];
#pragma unroll
      for (int off = 16; off > 0; off >>= 1) {
        const float ov = __shfl_xor(bestv, off, 32);
        const int   oi = __shfl_xor(besti, off, 32);
        if (ov > bestv || (ov == bestv && oi < besti)) { bestv = ov; besti = oi; }
      }
      if (tid == 0) { sLast = besti; ob[s] = besti; }
    }
    __syncthreads();
    last = sLast;
  }
}

// ---------------------------------------------------------------------------
extern "C" void kernel_launch(void* const* d_in, const int* in_sizes, int n_in,
                              void* d_out, int out_size, void* d_ws, size_t ws_size,
                              hipStream_t stream) {
  const float* pts   = (const float*)d_in[0];   // [2,8192,3]
  const float* feats = (const float*)d_in[1];   // [2,64,8192]
  // d_in[2] = npoint scalar (fixed 2048)
  float* dens = (float*)d_ws;                   // [2][8192] floats = 64 KB
  int*   out  = (int*)d_out;                    // [2][4096] int32 indices

  // 1024 waves total for density: 128 blocks x 256 threads (8 waves/block)
  density_kernel<<<128, 256, 0, stream>>>(pts, dens);
  // one workgroup per (batch, variant)
  fps_kernel<<<NBATCH * 2, 1024, 0, stream>>>(pts, feats, dens, out);
}